// SparseBottleneck_53515292508525
// MI455X (gfx1250) — compile-verified
//
#include <hip/hip_runtime.h>
#include <hip/hip_bf16.h>

typedef __attribute__((ext_vector_type(2))) float v2f;
typedef __attribute__((ext_vector_type(8))) float v8f;

// ---------------------------------------------------------------------------
// LDS weight layout: pair-interleaved  Wl[p][c][2]  with p = k/2, so a B
// fragment (W[kb][col], W[kb+1][col]) is one aligned ds_load_b64.
// A frag (16x4 f32): lanes 0-15 -> M=lane, VGPR0/1 = K=0/1 ; lanes 16-31 -> K=2/3
// B frag (4x16 f32): lanes 0-15 -> N=lane, VGPR0/1 = K=0/1 ; lanes 16-31 -> K=2/3
// C/D   (16x16 f32): VGPR r, lanes 0-15 -> M=r ; lanes 16-31 -> M=8+r
// ---------------------------------------------------------------------------

// out[N x C] = X[N x 64] @ W[64 x C] + bias ; grid-stride over 16x(16*CT) tiles
template <int C, int CT>
__global__ __launch_bounds__(256) void gemm_bias(const float* __restrict__ X,
                                                 const float* __restrict__ W,
                                                 const float* __restrict__ bias,
                                                 float* __restrict__ out,
                                                 int nrows) {
  constexpr int K = 64;
  constexpr int colGroups = C / (16 * CT);
  __shared__ float Wl[K * C];  // pair-interleaved
  for (int i4 = threadIdx.x; i4 < (K * C) / 4; i4 += 256) {
    const float4 v = ((const float4*)W)[i4];
    const int k = (i4 * 4) / C;
    const int c = (i4 * 4) % C;
    float* dst = &Wl[(k >> 1) * (2 * C) + (k & 1)];
    dst[(c + 0) * 2] = v.x;
    dst[(c + 1) * 2] = v.y;
    dst[(c + 2) * 2] = v.z;
    dst[(c + 3) * 2] = v.w;
  }
  __syncthreads();

  const int wave = threadIdx.x >> 5;
  const int lane = threadIdx.x & 31;
  const int r = lane & 15;
  const int khalf = (lane >> 4) << 1;  // 0 or 2
  const int cc = lane & 15;
  const long nTiles = (long)(nrows / 16) * colGroups;
  const long tstride = (long)gridDim.x * 8;

  for (long tile = (long)blockIdx.x * 8 + wave; tile < nTiles; tile += tstride) {
    const int rowTile = (int)(tile / colGroups);
    const int colGroup = (int)(tile % colGroups);
    const int row0 = rowTile * 16;
    const int col0base = colGroup * 16 * CT;
    const float* xrow = X + (size_t)(row0 + r) * K + khalf;

    const long nt = tile + tstride;  // prefetch next tile's A row
    if (nt < nTiles)
      __builtin_prefetch(X + (size_t)((int)(nt / colGroups) * 16 + r) * K, 0, 0);

    v8f acc[CT] = {};
#pragma unroll
    for (int k0 = 0; k0 < K; k0 += 4) {
      const v2f a = *(const v2f*)(xrow + k0);
      const int p = (k0 + khalf) >> 1;
#pragma unroll
      for (int t = 0; t < CT; ++t) {
        const v2f b = *(const v2f*)&Wl[p * (2 * C) + (col0base + t * 16 + cc) * 2];
        acc[t] = __builtin_amdgcn_wmma_f32_16x16x4_f32(false, a, false, b,
                                                       (short)0, acc[t], false, false);
      }
    }

    const int rbase = row0 + ((lane >> 4) << 3);
#pragma unroll
    for (int t = 0; t < CT; ++t) {
      const int col = col0base + t * 16 + cc;
      const float bv = bias[col];
#pragma unroll
      for (int rr = 0; rr < 8; ++rr)
        out[(size_t)(rbase + rr) * C + col] = acc[t][rr] + bv;
    }
  }
}

// ---------------------------------------------------------------------------
// Submanifold 3x3 conv: out = sum_k gather(H, nbrs[:,k]) @ w2[k] + b2
// All nine 64x64 weight matrices staged once in LDS (144 KB, pair-interleaved);
// the per-tile k-loop is then barrier-free. Invalid neighbors are handled
// branchlessly (clamped load * mask) so EXEC stays full for the WMMAs.
// ---------------------------------------------------------------------------
__global__ __launch_bounds__(256) void subm_gemm(const float* __restrict__ H,
                                                 const float* __restrict__ W2,
                                                 const float* __restrict__ b2,
                                                 const int* __restrict__ nbrs,
                                                 float* __restrict__ out,
                                                 int nrows) {
  constexpr int C = 64, K = 64;
  __shared__ float Wl[9 * K * C];  // 147456 bytes (WGP LDS = 320 KB)
  for (int i4 = threadIdx.x; i4 < (9 * K * C) / 4; i4 += 256) {
    const float4 v = ((const float4*)W2)[i4];
    const int kk = (i4 * 4) / C;     // global k-row 0..575
    const int c = (i4 * 4) % C;
    const int mat = kk >> 6, krow = kk & 63;
    float* dst = &Wl[mat * (K * C) + (krow >> 1) * (2 * C) + (krow & 1)];
    dst[(c + 0) * 2] = v.x;
    dst[(c + 1) * 2] = v.y;
    dst[(c + 2) * 2] = v.z;
    dst[(c + 3) * 2] = v.w;
  }
  __syncthreads();

  const int wave = threadIdx.x >> 5;
  const int lane = threadIdx.x & 31;
  const int r = lane & 15;
  const int khalf = (lane >> 4) << 1;
  const int cc = lane & 15;
  const long nRowTiles = nrows / 16;
  const long tstride = (long)gridDim.x * 8;

  for (long rowTile = (long)blockIdx.x * 8 + wave; rowTile < nRowTiles;
       rowTile += tstride) {
    const int row0 = (int)(rowTile * 16);
    v8f acc[4] = {};

    for (int k = 0; k < 9; ++k) {
      const int gidx = nbrs[(size_t)(row0 + r) * 9 + k];
      const float m = (gidx >= 0) ? 1.0f : 0.0f;
      const float* hrow = H + (size_t)(gidx >= 0 ? gidx : 0) * K + khalf;
      const float* wk = &Wl[k * (K * C)];
#pragma unroll
      for (int k0 = 0; k0 < K; k0 += 4) {
        v2f a = *(const v2f*)(hrow + k0);
        a.x *= m;
        a.y *= m;
        const int p = (k0 + khalf) >> 1;
#pragma unroll
        for (int t = 0; t < 4; ++t) {
          const v2f b = *(const v2f*)&wk[p * (2 * C) + (t * 16 + cc) * 2];
          acc[t] = __builtin_amdgcn_wmma_f32_16x16x4_f32(false, a, false, b,
                                                         (short)0, acc[t], false, false);
        }
      }
    }

    const int rbase = row0 + ((lane >> 4) << 3);
#pragma unroll
    for (int t = 0; t < 4; ++t) {
      const int col = t * 16 + cc;
      const float bv = b2[col];
#pragma unroll
      for (int rr = 0; rr < 8; ++rr)
        out[(size_t)(rbase + rr) * C + col] = acc[t][rr] + bv;
    }
  }
}

// ---------------------------------------------------------------------------
// BN statistics: per-channel sum / sum-of-squares with LDS reduce + f32 atomics
// ---------------------------------------------------------------------------
template <int C>
__global__ __launch_bounds__(256) void bn_stats(const float* __restrict__ T, int nrows,
                                                float* __restrict__ sum,
                                                float* __restrict__ sumsq) {
  constexpr int RPB = 256 / C;
  const int c = threadIdx.x % C;
  const int rsub = threadIdx.x / C;
  float s = 0.0f, q = 0.0f;
  for (long row = (long)blockIdx.x * RPB + rsub; row < nrows;
       row += (long)gridDim.x * RPB) {
    const float v = T[row * C + c];
    s += v;
    q += v * v;
  }
  __shared__ float ls[256], lq[256];
  ls[threadIdx.x] = s;
  lq[threadIdx.x] = q;
  __syncthreads();
  if (threadIdx.x < C) {
#pragma unroll
    for (int i = 1; i < RPB; ++i) {
      s += ls[threadIdx.x + i * C];
      q += lq[threadIdx.x + i * C];
    }
    atomicAdd(&sum[c], s);
    atomicAdd(&sumsq[c], q);
  }
}

__global__ void bn_finalize(const float* __restrict__ sum, const float* __restrict__ sumsq,
                            const float* __restrict__ g, const float* __restrict__ be,
                            float* __restrict__ scale, float* __restrict__ shift,
                            int C, float invN) {
  const int c = blockIdx.x * blockDim.x + threadIdx.x;
  if (c >= C) return;
  const float m = sum[c] * invN;
  const float var = sumsq[c] * invN - m * m;
  const float sc = g[c] * rsqrtf(var + 1e-5f);
  scale[c] = sc;
  shift[c] = be[c] - m * sc;
}

template <int C>
__global__ void bn_relu_apply(const float* __restrict__ T, const float* __restrict__ scale,
                              const float* __restrict__ shift, float* __restrict__ out,
                              long total) {
  const long i = ((long)blockIdx.x * blockDim.x + threadIdx.x) * 4;
  if (i >= total) return;
  const int c = (int)(i % C);
  const float4 v = *(const float4*)(T + i);
  float4 o;
  o.x = fmaxf(v.x * scale[c + 0] + shift[c + 0], 0.0f);
  o.y = fmaxf(v.y * scale[c + 1] + shift[c + 1], 0.0f);
  o.z = fmaxf(v.z * scale[c + 2] + shift[c + 2], 0.0f);
  o.w = fmaxf(v.w * scale[c + 3] + shift[c + 3], 0.0f);
  *(float4*)(out + i) = o;
}

// out = BN(ts) + relu(BN(t3));  t3 is in-place in d_out
__global__ void combine_kernel(const float* __restrict__ ts, float* __restrict__ t3,
                               const float* __restrict__ scs, const float* __restrict__ shs,
                               const float* __restrict__ sc3, const float* __restrict__ sh3,
                               long total) {
  const long i = ((long)blockIdx.x * blockDim.x + threadIdx.x) * 4;
  if (i >= total) return;
  const int c = (int)(i % 256);
  const float4 a = *(const float4*)(ts + i);
  const float4 b = *(const float4*)(t3 + i);
  float4 o;
  o.x = (a.x * scs[c + 0] + shs[c + 0]) + fmaxf(b.x * sc3[c + 0] + sh3[c + 0], 0.0f);
  o.y = (a.y * scs[c + 1] + shs[c + 1]) + fmaxf(b.y * sc3[c + 1] + sh3[c + 1], 0.0f);
  o.z = (a.z * scs[c + 2] + shs[c + 2]) + fmaxf(b.z * sc3[c + 2] + sh3[c + 2], 0.0f);
  o.w = (a.w * scs[c + 3] + shs[c + 3]) + fmaxf(b.w * sc3[c + 3] + sh3[c + 3], 0.0f);
  *(float4*)(t3 + i) = o;
}

__global__ void zero_kernel(float* __restrict__ p, int n) {
  const int i = blockIdx.x * blockDim.x + threadIdx.x;
  if (i < n) p[i] = 0.0f;
}

// ---------------------------------------------------------------------------
extern "C" void kernel_launch(void* const* d_in, const int* in_sizes, int n_in,
                              void* d_out, int out_size, void* d_ws, size_t ws_size,
                              hipStream_t stream) {
  const float* feats = (const float*)d_in[0];
  const int* nbrs = (const int*)d_in[1];
  const float* w1 = (const float*)d_in[2];
  const float* b1 = (const float*)d_in[3];
  const float* g1 = (const float*)d_in[4];
  const float* be1 = (const float*)d_in[5];
  const float* w2 = (const float*)d_in[6];
  const float* b2 = (const float*)d_in[7];
  const float* g2 = (const float*)d_in[8];
  const float* be2 = (const float*)d_in[9];
  const float* w3 = (const float*)d_in[10];
  const float* b3 = (const float*)d_in[11];
  const float* g3 = (const float*)d_in[12];
  const float* be3 = (const float*)d_in[13];
  const float* wsm = (const float*)d_in[14];
  const float* bs = (const float*)d_in[15];
  const float* gs = (const float*)d_in[16];
  const float* bes = (const float*)d_in[17];

  const long N = (long)in_sizes[0] / 64;  // 800000
  float* base = (float*)d_ws;
  float* bufA = base;             // N x 64 (t1 then t2)
  float* bufB = bufA + N * 64;    // N x 64 (h1 then h2)
  float* ts   = bufB + N * 64;    // N x 256 (skip-branch pre-BN)
  float* st   = ts + N * 256;     // stats region
  float* s1 = st;         float* q1 = s1 + 64;
  float* s2 = q1 + 64;    float* q2 = s2 + 64;
  float* s3 = q2 + 64;    float* q3 = s3 + 256;
  float* ssm = q3 + 256;  float* ssq = ssm + 256;
  float* sc1 = ssq + 256; float* sh1 = sc1 + 64;
  float* sc2 = sh1 + 64;  float* sh2 = sc2 + 64;
  float* sc3 = sh2 + 64;  float* sh3 = sc3 + 256;
  float* scs = sh3 + 256; float* shs = scs + 256;
  float* t3 = (float*)d_out;      // N x 256, produced in place

  const float invN = 1.0f / (float)N;
  const long el64 = N * 64, el256 = N * 256;

  // zero stat accumulators (4 layers x sum+sumsq = 1280 floats)
  zero_kernel<<<(1280 + 255) / 256, 256, 0, stream>>>(s1, 1280);

  // main branch GEMM1 and skip branch GEMM (grid-stride, persistent-ish blocks)
  gemm_bias<64, 4><<<1024, 256, 0, stream>>>(feats, w1, b1, bufA, (int)N);
  gemm_bias<256, 8><<<2048, 256, 0, stream>>>(feats, wsm, bs, ts, (int)N);

  bn_stats<64><<<4096, 256, 0, stream>>>(bufA, (int)N, s1, q1);
  bn_stats<256><<<4096, 256, 0, stream>>>(ts, (int)N, ssm, ssq);
  bn_finalize<<<1, 64, 0, stream>>>(s1, q1, g1, be1, sc1, sh1, 64, invN);
  bn_finalize<<<1, 256, 0, stream>>>(ssm, ssq, gs, bes, scs, shs, 256, invN);
  bn_relu_apply<64><<<(int)((el64 / 4 + 255) / 256), 256, 0, stream>>>(bufA, sc1, sh1, bufB, el64);

  // 3x3 submanifold gather-GEMM
  subm_gemm<<<1024, 256, 0, stream>>>(bufB, w2, b2, nbrs, bufA, (int)N);
  bn_stats<64><<<4096, 256, 0, stream>>>(bufA, (int)N, s2, q2);
  bn_finalize<<<1, 64, 0, stream>>>(s2, q2, g2, be2, sc2, sh2, 64, invN);
  bn_relu_apply<64><<<(int)((el64 / 4 + 255) / 256), 256, 0, stream>>>(bufA, sc2, sh2, bufB, el64);

  // GEMM3 into d_out, then final BN stats + residual combine
  gemm_bias<256, 8><<<2048, 256, 0, stream>>>(bufB, w3, b3, t3, (int)N);
  bn_stats<256><<<4096, 256, 0, stream>>>(t3, (int)N, s3, q3);
  bn_finalize<<<1, 256, 0, stream>>>(s3, q3, g3, be3, sc3, sh3, 256, invN);
  combine_kernel<<<(int)((el256 / 4 + 255) / 256), 256, 0, stream>>>(ts, t3, scs, shs, sc3, sh3, el256);
}